// sccnn_conv_no_n_to_n_sc_1_8976481648970
// MI455X (gfx1250) — compile-verified
//
#include <hip/hip_runtime.h>
#include <hip/hip_bf16.h>

typedef __attribute__((ext_vector_type(2))) float v2f;
typedef __attribute__((ext_vector_type(8))) float v8f;
typedef int v4i __attribute__((vector_size(16)));

#define N0d 3000
#define N1d 10000
#define F   16
#define WPB 8            // waves per block (wave32)
#define TPB (WPB * 32)
#define KC  128          // K rows of B staged in LDS per chunk

// ---- CDNA5 async global->LDS staging (guarded: falls back to reg staging) ----
#if __has_builtin(__builtin_amdgcn_global_load_async_to_lds_b128) && \
    __has_builtin(__builtin_amdgcn_s_wait_asynccnt)
#define USE_ASYNC_LDS 1
__device__ __forceinline__ void async_ld16(const void* g, void* l) {
    // GLOBAL_LOAD_ASYNC_TO_LDS_B128: memory -> LDS, no VGPR data, ASYNCcnt tracked
    __builtin_amdgcn_global_load_async_to_lds_b128(
        (__attribute__((address_space(1))) v4i*)g,
        (__attribute__((address_space(3))) v4i*)l,
        0, 0);
}
__device__ __forceinline__ void async_wait0() {
    __builtin_amdgcn_s_wait_asynccnt(0);
}
#else
#define USE_ASYNC_LDS 0
#endif

// ---------------- small elementwise kernels ----------------

__global__ void k_diag(const float* __restrict__ D, float* __restrict__ out, int n, int invert) {
    int i = blockIdx.x * blockDim.x + threadIdx.x;
    if (i < n) {
        float v = D[(size_t)i * n + i];
        out[i] = invert ? 1.0f / v : v;
    }
}

__global__ void k_rowscale(const float* __restrict__ x, const float* __restrict__ s,
                           float* __restrict__ y, int n) {
    int i = blockIdx.x * blockDim.x + threadIdx.x;   // over n*F
    if (i < n * F) y[i] = x[i] * s[i >> 4];
}

// O = c1*O*rs + c2*P   (rs and/or P may be null)
__global__ void k_epilogue(float* __restrict__ O, const float* __restrict__ P,
                           const float* __restrict__ rs, float c1, float c2, int n) {
    int i = blockIdx.x * blockDim.x + threadIdx.x;
    if (i < n * F) {
        float v = c1 * O[i];
        if (rs) v *= rs[i >> 4];
        if (P)  v += c2 * P[i];
        O[i] = v;
    }
}

// ---------------- split-K WMMA GEMM (accumulating) ----------------
// Computes O(+)= op(A) @ B for one K segment per grid.y slice.
// A element (m,k) at A + m*lda_m + k*lda_k  (symmetric L and B1^T read coalesced).
// B0/B1g: (K x 16) row-major.  DUAL=1 fuses two 16-col groups sharing the A fragment.
// AK1=1: lda_k==1 known at compile time -> single b64 A-fragment load.
template <int DUAL, int AK1>
__global__ __launch_bounds__(TPB) void k_gemm_acc(
    const float* __restrict__ A, long lda_m, long lda_k,
    int nrows, int K, int kseg,
    const float* __restrict__ B0, const float* __restrict__ B1g,
    float* __restrict__ O0, float* __restrict__ O1g)
{
    __shared__ float sB[(DUAL ? 2 : 1) * KC * F];
    float* sB0 = sB;
    float* sB1 = sB + KC * F;

    const int lane  = threadIdx.x & 31;
    const int wave  = threadIdx.x >> 5;
    const int tiles = (nrows + 15) >> 4;
    int tile = blockIdx.x * WPB + wave;
    const bool live = tile < tiles;
    if (tile >= tiles) tile = tiles - 1;           // clamp: keep EXEC full for WMMA

    const int kbeg = blockIdx.y * kseg;
    const int kend = min(K, kbeg + kseg);

    const int hl = lane >> 4;                      // half-wave: 0 or 1
    const int ln = lane & 15;
    int m = tile * 16 + ln;
    if (m >= nrows) m = nrows - 1;                 // clamp loads; store is guarded
    const float* Arow = A + (long)m * lda_m;

    v8f acc0 = {};
    v8f acc1 = {};

    for (int kb = kbeg; kb < kend; kb += KC) {
        const int kl = min(KC, kend - kb);         // multiple of 4 by construction
        // cooperative LDS staging of B chunk(s); rows >= kl are never read, skip them
        for (int t = threadIdx.x; t < (KC * F) / 4; t += TPB) {
            const int krow = t >> 2;               // 4 float4s per 16-wide row
            if (krow < kl) {
#if USE_ASYNC_LDS
                async_ld16(B0 + (size_t)kb * F + (size_t)t * 4, sB0 + t * 4);
                if (DUAL) async_ld16(B1g + (size_t)kb * F + (size_t)t * 4, sB1 + t * 4);
#else
                ((float4*)sB0)[t] = ((const float4*)(B0 + (size_t)kb * F))[t];
                if (DUAL) ((float4*)sB1)[t] = ((const float4*)(B1g + (size_t)kb * F))[t];
#endif
            }
        }
#if USE_ASYNC_LDS
        async_wait0();
#endif
        __syncthreads();

        const int ksteps = kl >> 2;
        for (int s = 0; s < ksteps; ++s) {
            const int  klo = s * 4 + hl * 2;
            const long kg  = (long)kb + klo;
            v2f a;                                  // A 16x4 fragment
            if (AK1) {
                const float2 av = *(const float2*)(Arow + kg);
                a.x = av.x; a.y = av.y;
            } else {
                a.x = Arow[kg * lda_k];
                a.y = Arow[(kg + 1) * lda_k];
            }
            v2f b;                                  // B 4x16 fragment
            b.x = sB0[klo * F + ln];
            b.y = sB0[(klo + 1) * F + ln];
            acc0 = __builtin_amdgcn_wmma_f32_16x16x4_f32(
                false, a, false, b, (short)0, acc0, false, false);
            if (DUAL) {
                v2f b1;
                b1.x = sB1[klo * F + ln];
                b1.y = sB1[(klo + 1) * F + ln];
                acc1 = __builtin_amdgcn_wmma_f32_16x16x4_f32(
                    false, a, false, b1, (short)0, acc1, false, false);
            }
        }
        __syncthreads();
    }

    if (live) {
        for (int r = 0; r < 8; ++r) {
            const int row = tile * 16 + r + hl * 8;  // C/D layout: vgpr r -> M=r / r+8
            if (row < nrows) {
                atomicAdd(&O0[(size_t)row * F + ln], acc0[r]);
                if (DUAL) atomicAdd(&O1g[(size_t)row * F + ln], acc1[r]);
            }
        }
    }
}

// ---------------- head: Y = relu( sum_k T_k @ W[:,:,k] ) ----------------
// T: nterms contiguous (n x 16) blocks. W: (16,16,KT) C-order.
__global__ __launch_bounds__(TPB) void k_head(
    const float* __restrict__ T, const float* __restrict__ W,
    int nterms, int KT, int n, float* __restrict__ Y)
{
    const int lane  = threadIdx.x & 31;
    const int wave  = threadIdx.x >> 5;
    const int tiles = (n + 15) >> 4;
    const int tile  = blockIdx.x * WPB + wave;
    if (tile >= tiles) return;                     // wave-uniform exit (no barriers here)

    const int hl = lane >> 4;
    const int ln = lane & 15;
    int m = tile * 16 + ln;
    if (m >= n) m = n - 1;

    v8f acc = {};
    for (int k = 0; k < nterms; ++k) {
        const float* Tk = T + (size_t)k * n * F;
        for (int s = 0; s < 4; ++s) {              // K = F_IN = 16 -> 4 WMMA steps
            const int i0 = s * 4 + hl * 2;
            const float2 av = *(const float2*)(Tk + (size_t)m * F + i0);
            v2f a; a.x = av.x; a.y = av.y;
            v2f b;
            b.x = W[(size_t)i0 * F * KT + (size_t)ln * KT + k];
            b.y = W[(size_t)(i0 + 1) * F * KT + (size_t)ln * KT + k];
            acc = __builtin_amdgcn_wmma_f32_16x16x4_f32(
                false, a, false, b, (short)0, acc, false, false);
        }
    }
    for (int r = 0; r < 8; ++r) {
        const int row = tile * 16 + r + hl * 8;
        if (row < n) Y[(size_t)row * F + ln] = fmaxf(acc[r], 0.0f);
    }
}

// ---------------- driver ----------------

extern "C" void kernel_launch(void* const* d_in, const int* in_sizes, int n_in,
                              void* d_out, int out_size, void* d_ws, size_t ws_size,
                              hipStream_t stream)
{
    (void)in_sizes; (void)n_in; (void)out_size; (void)ws_size;
    const float* x0 = (const float*)d_in[0];
    const float* x1 = (const float*)d_in[1];
    const float* B1 = (const float*)d_in[2];
    const float* L0 = (const float*)d_in[3];
    const float* L1 = (const float*)d_in[4];
    const float* D1 = (const float*)d_in[5];
    const float* D2 = (const float*)d_in[6];
    const float* W0 = (const float*)d_in[7];
    const float* W1 = (const float*)d_in[8];

    float* ws    = (float*)d_ws;
    float* d1inv = ws;                 // 3000
    float* d2v   = ws + 3000;          // 10000
    float* y0k   = ws + 13000;         // 48000  (d1inv * x0)
    float* t0    = ws + 61000;         // 4 x 48000 contiguous Chebyshev terms (node chain)
    float* t1    = t0 + 48000;
    float* t2    = t1 + 48000;
    float* t3    = t2 + 48000;
    float* u0    = ws + 253000;        // 7 x 160000 contiguous terms (edge chains)
    float* u1    = u0 + 160000;
    float* u2    = u1 + 160000;
    float* vc    = u2 + 160000;        // copy of x1 (term index 3)
    float* v1    = vc + 160000;
    float* v2    = v1 + 160000;
    float* v3    = v2 + 160000;

    float* y0 = (float*)d_out;         // (3000 x 16)
    float* y1 = y0 + (size_t)N0d * F;  // (10000 x 16)

    auto gemm = [&](const float* A, long ldam, long ldak, int nrows, int K, int segs,
                    const float* b0, const float* b1, float* o0, float* o1) {
        const int kseg  = (((K + segs - 1) / segs) + 3) & ~3;   // multiple of 4
        const int tiles = (nrows + 15) / 16;
        dim3 grid((tiles + WPB - 1) / WPB, (K + kseg - 1) / kseg);
        (void)hipMemsetAsync(o0, 0, (size_t)nrows * F * sizeof(float), stream);
        if (o1) (void)hipMemsetAsync(o1, 0, (size_t)nrows * F * sizeof(float), stream);
        if (b1) {
            k_gemm_acc<1, 0><<<grid, TPB, 0, stream>>>(A, ldam, ldak, nrows, K, kseg, b0, b1, o0, o1);
        } else if (ldak == 1) {
            k_gemm_acc<0, 1><<<grid, TPB, 0, stream>>>(A, ldam, ldak, nrows, K, kseg, b0, nullptr, o0, nullptr);
        } else {
            k_gemm_acc<0, 0><<<grid, TPB, 0, stream>>>(A, ldam, ldak, nrows, K, kseg, b0, nullptr, o0, nullptr);
        }
    };
    auto epi = [&](float* O, const float* P, const float* rs, float c1, float c2, int n) {
        k_epilogue<<<(n * F + 255) / 256, 256, 0, stream>>>(O, P, rs, c1, c2, n);
    };

    // diagonals + pre-scaled x0
    k_diag<<<(N0d + 255) / 256, 256, 0, stream>>>(D1, d1inv, N0d, 1);
    k_diag<<<(N1d + 255) / 256, 256, 0, stream>>>(D2, d2v, N1d, 0);
    k_rowscale<<<(N0d * F + 255) / 256, 256, 0, stream>>>(x0, d1inv, y0k, N0d);
    (void)hipMemcpyAsync(vc, x1, (size_t)N1d * F * sizeof(float), hipMemcpyDeviceToDevice, stream);

    // x0p = d1inv ⊙ (B1 @ x1)           A = B1 (row-major, lda_k==1)
    gemm(B1, (long)N1d, 1L, N0d, N1d, 16, x1, nullptr, t0, nullptr);
    epi(t0, nullptr, d1inv, 1.f, 0.f, N0d);

    // node Chebyshev chain (L0 symmetric -> read transposed for coalescing)
    gemm(L0, 1L, (long)N0d, N0d, N0d, 8, t0, nullptr, t1, nullptr);          // t1 = L0 t0
    gemm(L0, 1L, (long)N0d, N0d, N0d, 8, t1, nullptr, t2, nullptr);
    epi(t2, t0, nullptr, 2.f, -1.f, N0d);                                    // t2 = 2 L0 t1 - t0
    gemm(L0, 1L, (long)N0d, N0d, N0d, 8, t2, nullptr, t3, nullptr);
    epi(t3, t1, nullptr, 2.f, -1.f, N0d);                                    // t3 = 2 L0 t2 - t1

    // x1n = D2 ⊙ (B1^T @ (d1inv ⊙ x0))  A(m,k) = B1[k][m]
    gemm(B1, 1L, (long)N1d, N1d, N0d, 8, y0k, nullptr, u0, nullptr);
    epi(u0, nullptr, d2v, 1.f, 0.f, N1d);

    // fused edge Chebyshev chains over L1l (F=32 per pass: 5 -> 3 reads of the 400MB matrix)
    gemm(L1, 1L, (long)N1d, N1d, N1d, 8, u0, vc, u1, v1);                    // u1=L u0, v1=L x1
    gemm(L1, 1L, (long)N1d, N1d, N1d, 8, u1, v1, u2, v2);
    epi(u2, u0, nullptr, 2.f, -1.f, N1d);                                    // u2 = 2 L u1 - u0
    epi(v2, vc, nullptr, 2.f, -1.f, N1d);                                    // v2 = 2 L v1 - v0
    gemm(L1, 1L, (long)N1d, N1d, N1d, 8, v2, nullptr, v3, nullptr);
    epi(v3, v1, nullptr, 2.f, -1.f, N1d);                                    // v3 = 2 L v2 - v1

    // heads: relu(einsum) over contiguous term blocks
    k_head<<<(((N0d + 15) / 16) + WPB - 1) / WPB, TPB, 0, stream>>>(t0, W0, 4, 4, N0d, y0);
    k_head<<<(((N1d + 15) / 16) + WPB - 1) / WPB, TPB, 0, stream>>>(u0, W1, 7, 7, N1d, y1);
}